// Model_55645596287458
// MI455X (gfx1250) — compile-verified
//
#include <hip/hip_runtime.h>
#include <math.h>

typedef float v2f __attribute__((ext_vector_type(2)));
typedef float v8f __attribute__((ext_vector_type(8)));

#define THREADS 256
#define WAVES_PER_BLOCK (THREADS / 32)

// ---------------------------------------------------------------------------
// Prologue: R = Rx@Ry@Rz (row-major, ws[0..8]), folded translation
// c = -(T@R) (ws[9..11]). Single thread; negligible cost.
// ---------------------------------------------------------------------------
__global__ void pose_setup_kernel(const float* x, const float* y, const float* z,
                                  const float* roll, const float* pitch, const float* yaw,
                                  float* __restrict__ ws) {
  if (threadIdx.x != 0 || blockIdx.x != 0) return;
  float r = roll[0], p = pitch[0], yw = yaw[0];
  float cr = cosf(r),  sr = sinf(r);
  float cp = cosf(p),  sp = sinf(p);
  float cy = cosf(yw), sy = sinf(yw);
  float Rx[9] = {1.f, 0.f, 0.f,   0.f, cr, -sr,   0.f, sr,  cr};
  float Ry[9] = {cp,  0.f, sp,    0.f, 1.f, 0.f,  -sp, 0.f, cp};
  float Rz[9] = {cy, -sy, 0.f,    sy,  cy,  0.f,  0.f, 0.f, 1.f};
  float M[9], R[9];
  for (int i = 0; i < 3; ++i)
    for (int jj = 0; jj < 3; ++jj) {
      float s = 0.f;
      for (int k = 0; k < 3; ++k) s += Rx[i * 3 + k] * Ry[k * 3 + jj];
      M[i * 3 + jj] = s;
    }
  for (int i = 0; i < 3; ++i)
    for (int jj = 0; jj < 3; ++jj) {
      float s = 0.f;
      for (int k = 0; k < 3; ++k) s += M[i * 3 + k] * Rz[k * 3 + jj];
      R[i * 3 + jj] = s;
    }
  for (int i = 0; i < 9; ++i) ws[i] = R[i];
  float T0 = x[0], T1 = y[0], T2 = z[0];
  for (int jj = 0; jj < 3; ++jj)
    ws[9 + jj] = -(T0 * R[jj] + T1 * R[3 + jj] + T2 * R[6 + jj]);
}

// ---------------------------------------------------------------------------
// Main fused pass. Per wave iteration: 2 tiles x 16 points, each via one
// V_WMMA_F32_16X16X4_F32:
//   A (16x4): rows 0..2 = [R^T | c], rows 3..15 = 0     (wave-constant)
//   B (4x16): columns = [px, py, pz, 1]
//   D: VGPR0/1/2 lanes 0..15 = v.x / v.y / v.z, one point per lane.
// Masks use multiplied-through comparisons (no division): with vz>1,
//   u>1 <=> ph0>vz, u<1279 <=> ph0<1279*vz (same for w) — identical semantics.
// Per-thread partials -> wave butterfly -> LDS -> one deterministic partial
// per block (no float atomics, bit-stable across graph replays).
// ---------------------------------------------------------------------------
__device__ __forceinline__ float tile_body(const float* __restrict__ pts,
                                           float* __restrict__ out_v,
                                           v2f A, int base, int pidx, int comp,
                                           bool lowhalf, int lane, int npts) {
  int gidx = base + pidx;
  gidx = (gidx < npts) ? gidx : (npts - 1);
  const float* pp = pts + 3 * (size_t)gidx;
  float b0 = pp[comp];
  float ly = pp[1];
  float b1 = lowhalf ? ly : 1.0f;  // homogeneous 1 multiplies the c column
  v2f B;
  B.x = b0;
  B.y = b1;

  v8f C = {};
  v8f D = __builtin_amdgcn_wmma_f32_16x16x4_f32(
      /*neg_a=*/false, A, /*neg_b=*/false, B,
      /*c_mod=*/(short)0, C, /*reuse_a=*/false, /*reuse_b=*/false);

  float vx = D[0], vy = D[1], vz = D[2];

  // ph0 = FX*vx + CX*vz ; ph1 = FY*vy + CY*vz ; ph2 = vz
  float ph0 = fmaf(600.0f, vx, 640.0f * vz);
  float ph1 = fmaf(600.0f, vy, 360.0f * vz);
  // vz>1 implies vz>0, so divide-free bound checks are exact-equivalent:
  bool mask = (vz > 1.0f) & (vz < 10.0f) &
              (ph0 > vz) & (ph0 < 1279.0f * vz) &
              (ph1 > vz) & (ph1 < 719.0f * vz);
  const int gpt = base + lane;           // valid only for lanes < 16
  const bool active = lowhalf & (gpt < npts);
  mask = mask & active;

  float dn  = sqrtf(fmaf(vx, vx, fmaf(vy, vy, vz * vz)));
  float e   = (dn - 4.0f) * 0.5f;
  float obs = expf(-0.5f * e * e);

  if (active) {
    float* o = out_v + 3 * (size_t)gpt;
    o[0] = mask ? vx : 0.0f;
    o[1] = mask ? vy : 0.0f;
    o[2] = mask ? vz : 0.0f;
  }
  return mask ? obs : 0.0f;
}

__global__ __launch_bounds__(THREADS) void pose_obs_kernel(
    const float* __restrict__ pts, const float* __restrict__ ws,
    float* __restrict__ out_v, float* __restrict__ partials, int npts) {
  const int lane = threadIdx.x & 31;
  const int j = lane & 15;
  const bool lowhalf = (lane < 16);
  const bool arow = (j < 3);
  const int jc = arow ? j : 0;

  // A operand in the ISA 16x4 f32 layout (lanes 0-15: K0/K1, lanes 16-31: K2/K3)
  float a0 = lowhalf ? ws[jc] : ws[6 + jc];       // R[0][j] | R[2][j]
  float a1 = lowhalf ? ws[3 + jc] : ws[9 + jc];   // R[1][j] | c[j]
  a0 = arow ? a0 : 0.f;
  a1 = arow ? a1 : 0.f;
  v2f A;
  A.x = a0;
  A.y = a1;

  const int wave   = (int)((blockIdx.x * THREADS + threadIdx.x) >> 5);
  const int nwaves = (int)((gridDim.x * THREADS) >> 5);
  const int nchunks = (npts + 31) >> 5;  // 32 points (2 WMMA tiles) per chunk

  // B gather: lanes 0-15 load (x,y) of point `lane`; lanes 16-31 load z, b1=1.
  const int pidx = lowhalf ? lane : (lane - 16);
  const int comp = lowhalf ? 0 : 2;

  float acc = 0.f;
  for (int c = wave; c < nchunks; c += nwaves) {
    const int base = c << 5;
    // Two independent tiles: 4 loads + 2 WMMAs in flight per iteration.
    acc += tile_body(pts, out_v, A, base,      pidx, comp, lowhalf, lane, npts);
    acc += tile_body(pts, out_v, A, base + 16, pidx, comp, lowhalf, lane, npts);

    int nc = c + nwaves;
    if (nc < nchunks) {
      __builtin_prefetch(pts + 3 * (size_t)(nc << 5) + (size_t)lane * 12, 0, 1);
    }
  }

  // Wave butterfly reduction (wave32), then per-block partial (deterministic).
  for (int off = 16; off > 0; off >>= 1) acc += __shfl_xor(acc, off, 32);

  __shared__ float wsum[WAVES_PER_BLOCK];
  const int wid = threadIdx.x >> 5;
  if (lane == 0) wsum[wid] = acc;
  __syncthreads();
  if (threadIdx.x == 0) {
    float s = 0.f;
    for (int i = 0; i < WAVES_PER_BLOCK; ++i) s += wsum[i];
    partials[blockIdx.x] = s;
  }
}

// ---------------------------------------------------------------------------
// Finalize: fixed-order sum of block partials -> loss = 1/(sum + eps).
// ---------------------------------------------------------------------------
__global__ void finalize_kernel(const float* __restrict__ partials, int nblocks,
                                float* __restrict__ loss_out) {
  if (threadIdx.x == 0 && blockIdx.x == 0) {
    float s = 0.f;
    for (int i = 0; i < nblocks; ++i) s += partials[i];
    loss_out[0] = 1.0f / (s + 1e-6f);
  }
}

extern "C" void kernel_launch(void* const* d_in, const int* in_sizes, int n_in,
                              void* d_out, int out_size, void* d_ws, size_t ws_size,
                              hipStream_t stream) {
  const float* pts   = (const float*)d_in[0];
  const float* x     = (const float*)d_in[1];
  const float* y     = (const float*)d_in[2];
  const float* z     = (const float*)d_in[3];
  const float* roll  = (const float*)d_in[4];
  const float* pitch = (const float*)d_in[5];
  const float* yaw   = (const float*)d_in[6];

  const int npts = in_sizes[0] / 3;
  float* ws    = (float*)d_ws;               // ws[0..11] = R | c, ws[16..] partials
  float* out_v = (float*)d_out;              // npts*3 verts
  float* loss  = out_v + (size_t)npts * 3;   // +1 loss

  int nblocks = 1024;                        // 8192 waves; ~15 chunks each
  size_t cap = ws_size / sizeof(float);
  if (cap >= 17 && (size_t)nblocks + 16 > cap) nblocks = (int)(cap - 16);
  if (nblocks < 1) nblocks = 1;
  float* partials = ws + 16;

  pose_setup_kernel<<<1, 32, 0, stream>>>(x, y, z, roll, pitch, yaw, ws);
  pose_obs_kernel<<<nblocks, THREADS, 0, stream>>>(pts, ws, out_v, partials, npts);
  finalize_kernel<<<1, 32, 0, stream>>>(partials, nblocks, loss);
}